// Encoder_21620865368393
// MI455X (gfx1250) — compile-verified
//
#include <hip/hip_runtime.h>
#include <math.h>
#include <stdint.h>

#define IN_C   256
#define OUT_C  128
#define NGRAPH 64
#define ALPHA  0.1f
#define SCALEF 1.8f

typedef float v2f __attribute__((ext_vector_type(2)));
typedef float v8f __attribute__((ext_vector_type(8)));

// Types for the async-to-LDS builtin: param 0 is
// '__attribute__((vector_size(16))) int __device__ *' (AS1), param 1 the LDS
// (AS3) counterpart.
typedef int v4i_vs __attribute__((vector_size(16)));
typedef __attribute__((address_space(1))) v4i_vs* g_v4i_ptr;
typedef __attribute__((address_space(3))) v4i_vs* l_v4i_ptr;

#define XPAD 260   // 16B-aligned rows (260*4 % 16 == 0), bank-conflict free (260%64==4)

// ---------------------------------------------------------------------------
// Kernel 1: h = silu(x @ W^T + b); x_psc = h / max(||h||_2, 1e-12) * 1.8
// One block = 16 rows of x (full 128 output cols). 8 waves, one 16x16 WMMA
// tile per wave, K = 256 in steps of 4 using V_WMMA_F32_16X16X4_F32.
// x tile staged via GLOBAL_LOAD_ASYNC_TO_LDS_B128 when available.
// ---------------------------------------------------------------------------
__global__ __launch_bounds__(256) void gemm_silu_norm_k(
    const float* __restrict__ x, const float* __restrict__ W,
    const float* __restrict__ bias, float* __restrict__ x_psc, int N)
{
    __shared__ float xs[16 * XPAD];  // 16x256 x tile (padded)
    __shared__ float hs[16 * 130];   // 16x128 silu(h) tile (padded)
    __shared__ float rowsq[16];      // per-row sum of squares

    const int tid = threadIdx.x;
    const int m0  = blockIdx.x * 16;

#if defined(__gfx1250__) && __has_builtin(__builtin_amdgcn_global_load_async_to_lds_b128)
    // Async DMA the 16x256 tile straight into LDS (no VGPR round-trip).
    #pragma unroll
    for (int j = 0; j < 4; ++j) {
        int c   = tid + j * 256;        // float4 chunk id, 1024 total
        int r   = c >> 6;               // tile row
        int col = (c & 63) * 4;         // column within row
        int row = m0 + r; if (row > N - 1) row = N - 1;
        const float* gp = x + (size_t)row * IN_C + col;
        float*       lp = &xs[r * XPAD + col];
        __builtin_amdgcn_global_load_async_to_lds_b128(
            (g_v4i_ptr)(uintptr_t)gp, (l_v4i_ptr)(uintptr_t)lp, 0, 0);
    }
    if (tid < 16) rowsq[tid] = 0.0f;
    asm volatile("s_wait_asynccnt 0" ::: "memory");
#else
    #pragma unroll
    for (int j = 0; j < 16; ++j) {
        int idx = tid + j * 256;
        int r = idx >> 8, c = idx & 255;
        int row = m0 + r; if (row > N - 1) row = N - 1;
        xs[r * XPAD + c] = x[(size_t)row * IN_C + c];
    }
    if (tid < 16) rowsq[tid] = 0.0f;
#endif
    __syncthreads();

    const int lane = tid & 31;
    const int wave = tid >> 5;
    const int n0   = wave * 16;      // this wave's 16 output columns
    const int l    = lane & 15;
    const int half = lane >> 4;      // f32 WMMA A/B: lanes 16-31 hold K+2

    const float* wp = W  + (size_t)(n0 + l) * IN_C + 2 * half;  // B = W^T col
    const float* xp = xs + l * XPAD + 2 * half;                 // A row

    v8f acc = {};
    #pragma unroll
    for (int k = 0; k < IN_C; k += 4) {
        v2f a;  a.x  = xp[k]; a.y  = xp[k + 1];
        v2f bb; bb.x = wp[k]; bb.y = wp[k + 1];
        acc = __builtin_amdgcn_wmma_f32_16x16x4_f32(
                  false, a, false, bb, (short)0, acc, false, false);
    }

    // bias + SiLU, stage to LDS, accumulate row sum-of-squares.
    const float bv = bias[n0 + l];
    #pragma unroll
    for (int v = 0; v < 8; ++v) {
        float h = acc[v] + bv;
        float s = h / (1.0f + __expf(-h));          // SiLU
        int row = half * 8 + v;                     // C/D layout: M = 8*half+v
        hs[row * 130 + n0 + l] = s;
        atomicAdd(&rowsq[row], s * s);              // ds_add_f32
    }
    __syncthreads();

    // Normalize and write x_psc (coalesced).
    #pragma unroll
    for (int j = 0; j < 8; ++j) {
        int idx = tid + j * 256;
        int r = idx >> 7, c = idx & 127;
        if (m0 + r < N) {
            float sc = SCALEF / fmaxf(sqrtf(rowsq[r]), 1e-12f);
            x_psc[(size_t)(m0 + r) * OUT_C + c] = hs[r * 130 + c] * sc;
        }
    }
}

// ---------------------------------------------------------------------------
// Degree / norm pre-pass (GCN norm with self loops)
// ---------------------------------------------------------------------------
__global__ void deg_init_k(float* deg, int N) {
    int i = blockIdx.x * 256 + threadIdx.x;
    if (i < N) deg[i] = 1.0f;                        // self loop
}
__global__ void deg_edge_k(const int* __restrict__ dst, float* deg, int E) {
    int e = blockIdx.x * 256 + threadIdx.x;
    if (e < E) atomicAdd(&deg[dst[e]], 1.0f);
}
__global__ void dinv_k(float* deg, int N) {
    int i = blockIdx.x * 256 + threadIdx.x;
    if (i < N) deg[i] = rsqrtf(deg[i]);              // deg >= 1 always
}

// out = (0.9*dinv^2 + 0.1) * x_psc   (self-loop edge + teleport term folded)
__global__ void out_init_k(const float4* __restrict__ xp4,
                           const float*  __restrict__ dinv,
                           float4* __restrict__ out4, int N) {
    int idx = blockIdx.x * 256 + threadIdx.x;        // float4 index
    if (idx < N * (OUT_C / 4)) {
        int n = idx >> 5;                            // 32 float4 per node
        float di = dinv[n];
        float w = (1.0f - ALPHA) * di * di + ALPHA;
        float4 v = xp4[idx];
        v.x *= w; v.y *= w; v.z *= w; v.w *= w;
        out4[idx] = v;
    }
}

// ---------------------------------------------------------------------------
// Edge aggregation: one wave per 32 edges. Each lane loads one edge
// (coalesced); s/d/w broadcast to SGPRs via v_readlane so the x_psc gather
// and the out atomics use SADDR-form addressing (SGPR base + lane offset).
// out[dst] += 0.9 * dinv[src]*dinv[dst] * x_psc[src]
// ---------------------------------------------------------------------------
__global__ __launch_bounds__(256) void edge_agg_k(
    const int* __restrict__ src, const int* __restrict__ dst,
    const float* __restrict__ xp, const float* __restrict__ dinv,
    float* __restrict__ out, int E)
{
    const int lane = threadIdx.x & 31;
    const int wid  = (blockIdx.x * 256 + threadIdx.x) >> 5;
    const int base = wid * 32;
    if (base >= E) return;
    int ne = E - base; if (ne > 32) ne = 32;

    int myE = base + (lane < ne ? lane : 0);
    int s = src[myE];
    int d = dst[myE];
    float w = (1.0f - ALPHA) * dinv[s] * dinv[d];
    unsigned wbits = __float_as_uint(w);

    #pragma unroll
    for (int i = 0; i < 32; ++i) {
        if (i >= ne) break;                                   // wave-uniform
        int   si = __builtin_amdgcn_readlane(s, i);           // -> SGPR
        int   di = __builtin_amdgcn_readlane(d, i);           // -> SGPR
        float wi = __uint_as_float((unsigned)__builtin_amdgcn_readlane((int)wbits, i));
        const float4 v = ((const float4*)(xp + (size_t)si * OUT_C))[lane];
        float* o = out + (size_t)di * OUT_C + lane * 4;
        atomicAdd(o + 0, wi * v.x);
        atomicAdd(o + 1, wi * v.y);
        atomicAdd(o + 2, wi * v.z);
        atomicAdd(o + 3, wi * v.w);
    }
}

__global__ void zero_k(float* p, int n) {
    int i = blockIdx.x * 256 + threadIdx.x;
    if (i < n) p[i] = 0.0f;
}

// ---------------------------------------------------------------------------
// global_add_pool: per-block LDS [64 x 128] accumulator over a 1024-node
// chunk (batch is sorted -> few graphs per chunk), then flush to global.
// ---------------------------------------------------------------------------
__global__ __launch_bounds__(256) void pool_k(
    const float* __restrict__ out, const int* __restrict__ batch,
    float* __restrict__ xg, int N)
{
    __shared__ float acc[NGRAPH * OUT_C];            // 32 KB
    const int tid = threadIdx.x;
    #pragma unroll
    for (int j = 0; j < (NGRAPH * OUT_C) / 256; ++j) acc[tid + j * 256] = 0.0f;
    __syncthreads();

    const int CH = 1024;
    int base = blockIdx.x * CH;
    int lim  = base + CH; if (lim > N) lim = N;

    for (int i = 0; i < (CH * OUT_C) / 256; ++i) {
        int idx  = tid + i * 256;
        int node = base + (idx >> 7);
        if (node < lim) {
            int g = batch[node];
            int c = idx & 127;
            atomicAdd(&acc[g * OUT_C + c], out[(size_t)node * OUT_C + c]);
        }
    }
    __syncthreads();

    if (base < N) {
        int gmin = batch[base];
        int gmax = batch[lim - 1];
        int cnt  = (gmax - gmin + 1) * OUT_C;
        for (int idx2 = tid; idx2 < cnt; idx2 += 256) {
            int g = gmin + (idx2 >> 7);
            int c = idx2 & 127;
            atomicAdd(&xg[g * OUT_C + c], acc[g * OUT_C + c]);
        }
    }
}

// ---------------------------------------------------------------------------
extern "C" void kernel_launch(void* const* d_in, const int* in_sizes, int n_in,
                              void* d_out, int out_size, void* d_ws, size_t ws_size,
                              hipStream_t stream)
{
    const float* x     = (const float*)d_in[0];   // [N, 256]
    const int*   ei    = (const int*)  d_in[1];   // [2, E] flat
    const int*   batch = (const int*)  d_in[2];   // [N]
    const float* W     = (const float*)d_in[3];   // [128, 256]
    const float* b     = (const float*)d_in[4];   // [128]

    const int N = in_sizes[2];
    const int E = in_sizes[1] / 2;

    float* out = (float*)d_out;                   // [N, 128]
    float* xg  = out + (size_t)N * OUT_C;         // [64, 128]

    float* x_psc = (float*)d_ws;                  // [N, 128]
    float* dinv  = x_psc + (size_t)N * OUT_C;     // [N] (deg -> dinv in place)

    zero_k<<<(NGRAPH * OUT_C + 255) / 256, 256, 0, stream>>>(xg, NGRAPH * OUT_C);

    deg_init_k<<<(N + 255) / 256, 256, 0, stream>>>(dinv, N);
    deg_edge_k<<<(E + 255) / 256, 256, 0, stream>>>(ei + E, dinv, E);
    dinv_k    <<<(N + 255) / 256, 256, 0, stream>>>(dinv, N);

    gemm_silu_norm_k<<<(N + 15) / 16, 256, 0, stream>>>(x, W, b, x_psc, N);

    int q4 = N * (OUT_C / 4);
    out_init_k<<<(q4 + 255) / 256, 256, 0, stream>>>(
        (const float4*)x_psc, dinv, (float4*)out, N);

    int waves = (E + 31) / 32;
    edge_agg_k<<<(waves + 7) / 8, 256, 0, stream>>>(ei, ei + E, x_psc,
                                                    dinv, out, E);

    pool_k<<<(N + 1023) / 1024, 256, 0, stream>>>(out, batch, xg, N);
}